// MedianBlur_46136538693668
// MI455X (gfx1250) — compile-verified
//
#include <hip/hip_runtime.h>
#include <hip/hip_bf16.h>

// MedianBlur 3x3, zero padding, input [8,16,512,512] f32 -> same shape f32.
// Strategy: per-block 128x16 tile staged into LDS (with halo) via CDNA5 async
// global->LDS copies (ASYNCcnt + s_wait_asynccnt), med3-network median-of-9.

#define IMG_H 512
#define IMG_W 512
#define TILE_W 128
#define TILE_H 16
#define LDSW   136        // floats per LDS row: 4 halo left + 128 + 4 halo right
#define NROWS  18         // TILE_H + 2 halo rows
#define NCHUNK 34         // LDSW / 4 float4 chunks per row
#define NPLANES (8 * 16)

#if defined(__HIP_DEVICE_COMPILE__) && __has_builtin(__builtin_amdgcn_global_load_async_to_lds_b128)
#define USE_ASYNC_LDS 1
// Pointer types matching the builtin signature: (v4i as1*, v4i as3*, imm, imm)
typedef int v4i __attribute__((vector_size(16)));
typedef __attribute__((address_space(1))) v4i* gptr_v4i;
typedef __attribute__((address_space(3))) v4i* lptr_v4i;
#endif

__device__ __forceinline__ float med3f(float a, float b, float c) {
#if defined(__HIP_DEVICE_COMPILE__) && __has_builtin(__builtin_amdgcn_fmed3f)
    return __builtin_amdgcn_fmed3f(a, b, c);      // v_med3_f32
#else
    return fmaxf(fminf(a, b), fminf(fmaxf(a, b), c));
#endif
}
__device__ __forceinline__ float min3f(float a, float b, float c) { return fminf(fminf(a, b), c); }
__device__ __forceinline__ float max3f(float a, float b, float c) { return fmaxf(fmaxf(a, b), c); }

// Median of 9 via the classic 3-input-op network (13 VALU ops):
// sort each row of 3, then med3(max of mins, med of meds, min of maxs).
__device__ __forceinline__ float med9(const float p[9]) {
    float lo0 = min3f(p[0], p[1], p[2]);
    float md0 = med3f(p[0], p[1], p[2]);
    float hi0 = max3f(p[0], p[1], p[2]);
    float lo1 = min3f(p[3], p[4], p[5]);
    float md1 = med3f(p[3], p[4], p[5]);
    float hi1 = max3f(p[3], p[4], p[5]);
    float lo2 = min3f(p[6], p[7], p[8]);
    float md2 = med3f(p[6], p[7], p[8]);
    float hi2 = max3f(p[6], p[7], p[8]);
    float mn  = max3f(lo0, lo1, lo2);
    float md  = med3f(md0, md1, md2);
    float mx  = min3f(hi0, hi1, hi2);
    return med3f(mn, md, mx);
}

__global__ __launch_bounds__(256)
void MedianBlur_46136538693668_kernel(const float* __restrict__ in, float* __restrict__ out) {
    __shared__ __attribute__((aligned(16))) float tile[NROWS * LDSW];

    const int tid = threadIdx.x;
    const int c0  = blockIdx.x * TILE_W;          // tile origin col
    const int r0  = blockIdx.y * TILE_H;          // tile origin row
    const long long base = (long long)blockIdx.z * (IMG_H * IMG_W);
    const float* plane = in + base;

    // ---- Stage (TILE_H+2) x (TILE_W+8) halo region into LDS -----------------
    // LDS col c corresponds to global col (c0 - 4 + c); LDS row r -> (r0 - 1 + r).
    // Addresses are clamped into the image so async loads are always in-bounds;
    // clamped (garbage) cells are never consumed (border taps are zero-substituted).
    for (int i = tid; i < NROWS * NCHUNK; i += 256) {
        int r  = i / NCHUNK;
        int ck = i - r * NCHUNK;
        int gr = r0 - 1 + r;
        gr = gr < 0 ? 0 : (gr > IMG_H - 1 ? IMG_H - 1 : gr);
        int gc = c0 - 4 + (ck << 2);
        gc = gc < 0 ? 0 : (gc > IMG_W - 4 ? IMG_W - 4 : gc);
        const float* gsrc = plane + gr * IMG_W + gc;
        float* ldst = &tile[r * LDSW + (ck << 2)];
#if defined(USE_ASYNC_LDS)
        // CDNA5 async DMA-style copy: global_load_async_to_lds_b128 (ASYNCcnt)
        __builtin_amdgcn_global_load_async_to_lds_b128(
            (gptr_v4i)gsrc, (lptr_v4i)ldst, /*offset=*/0, /*cpol=*/0);
#else
        *(float4*)ldst = *(const float4*)gsrc;
#endif
    }
#if defined(USE_ASYNC_LDS)
#if __has_builtin(__builtin_amdgcn_s_wait_asynccnt)
    __builtin_amdgcn_s_wait_asynccnt(0);
#else
    asm volatile("s_wait_asynccnt 0" ::: "memory");
#endif
#endif
    __syncthreads();

    // ---- Compute: each thread -> 4 wide x 2 tall micro-tile -----------------
    const int lane  = tid & 31;        // 32 lanes * 4 px = 128 cols
    const int rowg  = tid >> 5;        // 8 row-groups * 2 rows = 16 rows
    const int x0    = lane << 2;       // tile-local col of first output
    const int ybase = rowg << 1;       // tile-local row of first output

    // Pull 4 LDS rows x 12 floats (covers both output rows' 3x3 windows).
    float wv[4][12];
#pragma unroll
    for (int r = 0; r < 4; ++r) {
        const float* lp = &tile[(ybase + r) * LDSW + x0];
        float4 a = *(const float4*)(lp);
        float4 b = *(const float4*)(lp + 4);
        float4 c = *(const float4*)(lp + 8);
        wv[r][0] = a.x; wv[r][1] = a.y; wv[r][2]  = a.z; wv[r][3]  = a.w;
        wv[r][4] = b.x; wv[r][5] = b.y; wv[r][6]  = b.z; wv[r][7]  = b.w;
        wv[r][8] = c.x; wv[r][9] = c.y; wv[r][10] = c.z; wv[r][11] = c.w;
    }

    const bool interior = (r0 > 0) && (r0 + TILE_H < IMG_H) &&
                          (c0 > 0) && (c0 + TILE_W < IMG_W);

    float* outplane = out + base;
#pragma unroll
    for (int k = 0; k < 2; ++k) {
        const int gy = r0 + ybase + k;
        float res[4];
#pragma unroll
        for (int j = 0; j < 4; ++j) {
            float p[9];
#pragma unroll
            for (int i = 0; i < 3; ++i)
#pragma unroll
                for (int d = 0; d < 3; ++d)
                    p[i * 3 + d] = wv[k + i][3 + j + d];
            if (!interior) {
                const int gx = c0 + x0 + j;
#pragma unroll
                for (int i = 0; i < 3; ++i) {
                    const int yy = gy - 1 + i;
#pragma unroll
                    for (int d = 0; d < 3; ++d) {
                        const int xx = gx - 1 + d;
                        if (yy < 0 || yy >= IMG_H || xx < 0 || xx >= IMG_W)
                            p[i * 3 + d] = 0.0f;
                    }
                }
            }
            res[j] = med9(p);
        }
        float4 o; o.x = res[0]; o.y = res[1]; o.z = res[2]; o.w = res[3];
        *(float4*)(outplane + gy * IMG_W + c0 + x0) = o;   // global_store_b128
    }
}

extern "C" void kernel_launch(void* const* d_in, const int* in_sizes, int n_in,
                              void* d_out, int out_size, void* d_ws, size_t ws_size,
                              hipStream_t stream) {
    const float* x  = (const float*)d_in[0];
    float* out      = (float*)d_out;
    dim3 grid(IMG_W / TILE_W, IMG_H / TILE_H, NPLANES);   // 4 x 32 x 128 blocks
    dim3 block(256);                                       // 8 wave32 per block
    hipLaunchKernelGGL(MedianBlur_46136538693668_kernel, grid, block, 0, stream, x, out);
}